// ParallelWHResNet_80161269613222
// MI455X (gfx1250) — compile-verified
//
#include <hip/hip_runtime.h>
#include <hip/hip_bf16.h>
#include <math.h>

typedef __attribute__((ext_vector_type(2))) float v2f;
typedef __attribute__((ext_vector_type(8))) float v8f;

constexpr int Bn  = 256;
constexpr int Tn  = 8192;
constexpr int Ln  = 16;
constexpr int Cn  = Tn / Ln;        // 512 chunks
constexpr int BTn = Bn * Tn;        // 2,097,152
constexpr int NF  = 5;              // G1o0, G1o1, G2i0, G2i1, G3
constexpr int WT_STRIDE = 24 * 16;  // packed (6 ksteps x 2 half x 16 n x 2 j)
constexpr int GY_OFF    = NF * WT_STRIDE;       // 1920
constexpr int DATA_OFF  = 4096;                 // params region padded to 16KB
constexpr long long TL_OFF    = DATA_OFF + 3LL * BTn;
constexpr long long TL_STRIDE = (long long)Bn * Cn * 6;

// ---------------------------------------------------------------------------
// Setup: build per-filter boundary matrices from ARMA coefficients.
// Wb packed layout for b64 B-fragment loads:
//   Wb[((ks*2 + half)*16 + n)*2 + j] = d y_chunk[n] / d x_win[w],  w = ks*4+half*2+j
// Window slot w corresponds to input time t0 - wpad + w; wpad = 4 (nb=4
// filters) or 8 (G3, nb=6: two oldest columns are exactly zero, which keeps
// the high edge of the window inside [0, Tn) for every chunk).
// Gy[r][j]: d y_chunk[r] / d y[-1-j] (transient Jacobian).
// ---------------------------------------------------------------------------
__global__ void setup_kernel(const float* G1_b, const float* G1_a,
                             const float* G2_b, const float* G2_a,
                             const float* G3_b, const float* G3_a,
                             float* ws) {
  int f = threadIdx.x;
  if (f >= NF) return;
  float bco[6], aco[6];
  int nb, na;
  if (f < 2)      { nb = na = 4; for (int k = 0; k < 4; ++k) { bco[k] = G1_b[f*4+k]; aco[k] = G1_a[f*4+k]; } }
  else if (f < 4) { nb = na = 4; int i = f - 2;
                    for (int k = 0; k < 4; ++k) { bco[k] = G2_b[i*4+k]; aco[k] = G2_a[i*4+k]; } }
  else            { nb = na = 6; for (int k = 0; k < 6; ++k) { bco[k] = G3_b[k]; aco[k] = G3_a[k]; } }
  const int wpad   = (f == 4) ? 8 : 4;
  const int w_used = (f == 4) ? 24 : 20;   // ksteps*4

  float* Wb = ws + f * WT_STRIDE;
  for (int w = 0; w < 24; ++w) {
    // exact Jacobian column: impulse at window slot w (time s0 = w - wpad),
    // pre-chunk y forced to zero, N_K=1 input delay.
    float K[16];
    const int s0 = w - wpad;
    for (int r = 0; r < 16; ++r) {
      float v = 0.f;
      const int kk = r - 1 - s0;
      if (w < w_used && kk >= 0 && kk < nb) v = bco[kk];
      for (int j = 0; j < na; ++j) { int s = r - 1 - j; if (s >= 0) v -= aco[j] * K[s]; }
      K[r] = v;
    }
    const int ks = w >> 2, hf = (w >> 1) & 1, j = w & 1;
    for (int n = 0; n < 16; ++n)
      Wb[((ks * 2 + hf) * 16 + n) * 2 + j] = (w < w_used) ? K[n] : 0.f;
  }

  // Transient matrix: impulse on y[-1-j], zero input.
  float* Gy = ws + GY_OFF + f * 96;
  for (int j = 0; j < 6; ++j) {
    float g[16];
    for (int r = 0; r < 16; ++r) {
      float v = 0.f;
      if (j < na) {
        for (int m = 0; m < na; ++m) {
          int s = r - 1 - m;
          float yv = (s >= 0) ? g[s] : ((s == -1 - j) ? 1.f : 0.f);
          v -= aco[m] * yv;
        }
      }
      g[r] = v;
    }
    for (int r = 0; r < 16; ++r) Gy[r * 6 + j] = g[r];
  }
}

// ---------------------------------------------------------------------------
// WMMA fragments (f32 16x16x4): A lane L -> M=L&15 (batch), K=2*(L>>4)+j ;
// B lane -> K=2*(L>>4)+j, N=L&15 (time) ; C/D lane -> M=v+8*(L>>4), N=L&15.
// ---------------------------------------------------------------------------
template <int KSTEPS>
__device__ __forceinline__ void load_A_frags(v2f (&a)[KSTEPS],
                                             const float* __restrict__ src,
                                             int rowBase, int t0, int wpad, int lane) {
  const int m  = lane & 15;
  const int hf = lane >> 4;
  const float* row = src + (size_t)(rowBase + m) * Tn;
  const int base = t0 - wpad + hf * 2;
  if (t0 >= wpad) {                 // wave-uniform: hot path, 511/512 chunks
#pragma unroll
    for (int ks = 0; ks < KSTEPS; ++ks)
      a[ks] = *(const v2f*)(row + base + ks * 4);   // aligned b64, imm offsets
  } else {                          // chunk 0 only: zero initial conditions
#pragma unroll
    for (int ks = 0; ks < KSTEPS; ++ks) {
#pragma unroll
      for (int j = 0; j < 2; ++j) {
        const int t = base + ks * 4 + j;
        const float v = row[t < 0 ? 0 : t];         // clamped addr, no branch
        a[ks][j] = (t < 0) ? 0.f : v;
      }
    }
  }
}

template <int KSTEPS>
__device__ __forceinline__ v8f wmma_acc(const v2f (&a)[KSTEPS],
                                        const float* __restrict__ Wb, int lane) {
  const int n  = lane & 15;
  const int hf = lane >> 4;
  const float* bp = Wb + ((size_t)hf * 16 + n) * 2;
  v8f acc = {};
#pragma unroll
  for (int ks = 0; ks < KSTEPS; ++ks) {
    v2f b = *(const v2f*)(bp + ks * 64);            // packed pair, b64 load
    acc = __builtin_amdgcn_wmma_f32_16x16x4_f32(
        false, a[ks], false, b, (short)0, acc, false, false);
  }
  return acc;
}

__device__ __forceinline__ void store_tile(float* __restrict__ dst, v8f acc,
                                           int rowBase, int t0, int lane) {
  const int n = lane & 15, hf = lane >> 4;
#pragma unroll
  for (int v = 0; v < 8; ++v) {
    const int m = v + hf * 8;
    dst[(size_t)(rowBase + m) * Tn + t0 + n] = acc[v];   // contiguous along t
  }
}

__device__ __forceinline__ void store_tails(float* __restrict__ tl, v8f acc,
                                            int rowBase, int cidx, int na, int lane) {
  const int n = lane & 15, hf = lane >> 4;
  if (n >= 16 - na) {
    const int j = 15 - n;                                // tail[j] = y[end-1-j]
#pragma unroll
    for (int v = 0; v < 8; ++v) {
      const int m = v + hf * 8;
      tl[((size_t)(rowBase + m) * Cn + cidx) * 6 + j] = acc[v];
    }
  }
}

// ---- Stage 1: G1 forced responses (A fragments shared by both channels) -----
__global__ void forced1_kernel(const float* __restrict__ u, float* ws) {
  const int lane = threadIdx.x;
  const int c  = blockIdx.x * 4 + threadIdx.y;
  const int rowBase = blockIdx.y * 16;
  const int t0 = c * Ln;
  v2f au[5];
  load_A_frags<5>(au, u, rowBase, t0, 4, lane);
  v8f d0 = wmma_acc<5>(au, ws,             lane);
  v8f d1 = wmma_acc<5>(au, ws + WT_STRIDE, lane);
  float* F1_0 = ws + DATA_OFF;
  float* F1_1 = F1_0 + BTn;
  store_tile(F1_0, d0, rowBase, t0, lane);
  store_tile(F1_1, d1, rowBase, t0, lane);
  store_tails(ws + TL_OFF,             d0, rowBase, c, 4, lane);
  store_tails(ws + TL_OFF + TL_STRIDE, d1, rowBase, c, 4, lane);
}

// ---- Sequential chunk-tail scan: 512 steps of a tiny na-dim affine map ------
// In-place: slot c is replaced by the INCOMING tail for chunk c (= tail_{c-1}).
__global__ void tails_kernel(float* ws, int fbase, int nfilt) {
  const int gid = blockIdx.x * blockDim.x + threadIdx.x;
  if (gid >= nfilt * Bn) return;
  const int f = fbase + gid / Bn;
  const int b = gid % Bn;
  const int na = (f == 4) ? 6 : 4;
  const float* Gy = ws + GY_OFF + f * 96;
  float* tl = ws + TL_OFF + (long long)f * TL_STRIDE + (long long)b * Cn * 6;
  float tail[6] = {0.f, 0.f, 0.f, 0.f, 0.f, 0.f};
  for (int c = 0; c < Cn; ++c) {
    float ft[6], nt[6];
    for (int j = 0; j < na; ++j) ft[j] = tl[c * 6 + j];
    for (int i = 0; i < na; ++i) {
      float v = ft[i];
      const int r = 15 - i;
      for (int j = 0; j < na; ++j) v += Gy[r * 6 + j] * tail[j];
      nt[i] = v;
    }
    for (int j = 0; j < na; ++j) tl[c * 6 + j] = tail[j];  // incoming tail
    for (int j = 0; j < na; ++j) tail[j] = nt[j];
  }
}

// ---- Fused transient-add + channelwise tanh MLP (writes z over F1 in place) -
__global__ void nl_kernel(float* ws, const float* __restrict__ w1,
                          const float* __restrict__ b1,
                          const float* __restrict__ w2,
                          const float* __restrict__ b2) {
  const int idx = blockIdx.x * blockDim.x + threadIdx.x;
  if (idx >= BTn) return;
  const int b = idx / Tn, t = idx % Tn, c = t >> 4, r = t & 15;
  for (int ch = 0; ch < 2; ++ch) {
    float* Fc = ws + DATA_OFF + (size_t)ch * BTn;
    const float* Gy  = ws + GY_OFF + ch * 96;
    const float* Tin = ws + TL_OFF + (long long)ch * TL_STRIDE + ((long long)b * Cn + c) * 6;
    float y = Fc[idx];
    for (int j = 0; j < 4; ++j) y += Gy[r * 6 + j] * Tin[j];
    float acc = b2[ch];
    for (int h = 0; h < 10; ++h)
      acc += w2[ch * 10 + h] * tanhf(y * w1[ch * 10 + h] + b1[ch * 10 + h]);
    Fc[idx] = acc;
  }
}

// ---- Stage 2: G2 (two input channels) + G3 (residual branch) forced ---------
__global__ void forced2_kernel(const float* __restrict__ u, float* ws) {
  const int lane = threadIdx.x;
  const int c  = blockIdx.x * 4 + threadIdx.y;
  const int rowBase = blockIdx.y * 16;
  const int t0 = c * Ln;
  const float* Z0 = ws + DATA_OFF;
  const float* Z1 = Z0 + BTn;
  v2f a2[5], a3[5], a4[6];
  load_A_frags<5>(a2, Z0, rowBase, t0, 4, lane);
  load_A_frags<5>(a3, Z1, rowBase, t0, 4, lane);
  load_A_frags<6>(a4, u,  rowBase, t0, 8, lane);
  v8f d2 = wmma_acc<5>(a2, ws + 2 * WT_STRIDE, lane);
  v8f d3 = wmma_acc<5>(a3, ws + 3 * WT_STRIDE, lane);
  v8f d4 = wmma_acc<6>(a4, ws + 4 * WT_STRIDE, lane);
  float* F2 = ws + DATA_OFF + 2 * (size_t)BTn;
  v8f s = d2 + d3 + d4;
  store_tile(F2, s, rowBase, t0, lane);
  store_tails(ws + TL_OFF + 2 * TL_STRIDE, d2, rowBase, c, 4, lane);
  store_tails(ws + TL_OFF + 3 * TL_STRIDE, d3, rowBase, c, 4, lane);
  store_tails(ws + TL_OFF + 4 * TL_STRIDE, d4, rowBase, c, 6, lane);
}

// ---- Final: add transients of G2/G3 filters to the summed forced response ---
__global__ void final_kernel(float* ws, float* __restrict__ out) {
  const int idx = blockIdx.x * blockDim.x + threadIdx.x;
  if (idx >= BTn) return;
  const int b = idx / Tn, t = idx % Tn, c = t >> 4, r = t & 15;
  float v = ws[DATA_OFF + 2 * (size_t)BTn + idx];
  for (int f = 2; f < 5; ++f) {
    const int na = (f == 4) ? 6 : 4;
    const float* Gy  = ws + GY_OFF + f * 96;
    const float* Tin = ws + TL_OFF + (long long)f * TL_STRIDE + ((long long)b * Cn + c) * 6;
    for (int j = 0; j < na; ++j) v += Gy[r * 6 + j] * Tin[j];
  }
  out[idx] = v;
}

extern "C" void kernel_launch(void* const* d_in, const int* in_sizes, int n_in,
                              void* d_out, int out_size, void* d_ws, size_t ws_size,
                              hipStream_t stream) {
  const float* u    = (const float*)d_in[0];
  const float* G1_b = (const float*)d_in[1];
  const float* G1_a = (const float*)d_in[2];
  const float* w1   = (const float*)d_in[3];
  const float* b1   = (const float*)d_in[4];
  const float* w2   = (const float*)d_in[5];
  const float* b2   = (const float*)d_in[6];
  const float* G2_b = (const float*)d_in[7];
  const float* G2_a = (const float*)d_in[8];
  const float* G3_b = (const float*)d_in[9];
  const float* G3_a = (const float*)d_in[10];
  float* ws  = (float*)d_ws;
  float* out = (float*)d_out;

  setup_kernel<<<1, 8, 0, stream>>>(G1_b, G1_a, G2_b, G2_a, G3_b, G3_a, ws);
  forced1_kernel<<<dim3(Cn / 4, Bn / 16), dim3(32, 4), 0, stream>>>(u, ws);
  tails_kernel<<<(2 * Bn + 127) / 128, 128, 0, stream>>>(ws, 0, 2);
  nl_kernel<<<(BTn + 255) / 256, 256, 0, stream>>>(ws, w1, b1, w2, b2);
  forced2_kernel<<<dim3(Cn / 4, Bn / 16), dim3(32, 4), 0, stream>>>(u, ws);
  tails_kernel<<<(3 * Bn + 127) / 128, 128, 0, stream>>>(ws, 2, 3);
  final_kernel<<<(BTn + 255) / 256, 256, 0, stream>>>(ws, out);
}